// BERTMADActQuantizer_47691316855355
// MI455X (gfx1250) — compile-verified
//
#include <hip/hip_runtime.h>

// Elementwise MAD-group quantize/dequantize. Memory-bound: 512 MiB of HBM
// traffic -> ~23 us floor at 23.3 TB/s. Strategy: 128-bit NT streaming
// loads/stores, scalar-register tables, branchless sorted-select for the
// per-element group gather, exact f32 divide + round-half-even to match the
// JAX reference numerically.

typedef float v4f __attribute__((ext_vector_type(4)));

namespace {
constexpr int G = 10;      // number of MAD groups (static in reference)
constexpr int BLOCK = 256; // 8 waves (wave32)
constexpr int ITERS = 8;   // float4 iterations per thread (amortize setup)
}

__device__ __forceinline__ float madq1(float x,
                                       const float (&m)[G],
                                       const float (&d)[G],
                                       const float (&z)[G]) {
  const float ax = __builtin_fabsf(x);
  // medians are sorted ascending, so
  //   grp = clip(sum_i(ax >= m[i]), 0, G-1)
  //   tbl[grp] == chain: start tbl[0]; for k=1..G-1: if ax >= m[k-1] -> tbl[k]
  float dd = d[0];
  float zz = z[0];
#pragma unroll
  for (int k = 1; k < G; ++k) {
    const bool ge = ax >= m[k - 1];   // v_cmp (abs modifier folds in)
    dd = ge ? d[k] : dd;              // v_cndmask
    zz = ge ? z[k] : zz;              // v_cndmask
  }
  // round-half-to-even matches jnp.round; exact IEEE divide matches x / d
  float q = __builtin_rintf(x / dd) + zz;
  q = __builtin_amdgcn_fmed3f(q, 0.0f, 255.0f); // clip(., 0, N_LEVELS-1)
  return (q - zz) * dd;
}

__global__ __launch_bounds__(BLOCK) void madq_kernel(
    const float* __restrict__ x, const float* __restrict__ medians,
    const float* __restrict__ deltas, const float* __restrict__ zps,
    float* __restrict__ out, int numV4, int n) {
  // Uniform-address table loads -> SGPRs (s_load), amortized over the loop.
  float m[G], d[G], z[G];
#pragma unroll
  for (int i = 0; i < G; ++i) {
    m[i] = medians[i];
    d[i] = deltas[i];
    z[i] = zps[i];
  }

  const v4f* __restrict__ x4 = reinterpret_cast<const v4f*>(x);
  v4f* __restrict__ o4 = reinterpret_cast<v4f*>(out);

  const int tid = blockIdx.x * BLOCK + threadIdx.x;
  const int stride = gridDim.x * BLOCK;

  for (int i = tid; i < numV4; i += stride) {
    if (i + stride < numV4) {
      // gfx1250 global_prefetch_b8 of this thread's next line (streaming).
      __builtin_prefetch(reinterpret_cast<const char*>(x4 + i + stride), 0, 0);
    }
    v4f v = __builtin_nontemporal_load(x4 + i);   // global_load_b128, TH=NT
    v4f r;
    r.x = madq1(v.x, m, d, z);
    r.y = madq1(v.y, m, d, z);
    r.z = madq1(v.z, m, d, z);
    r.w = madq1(v.w, m, d, z);
    __builtin_nontemporal_store(r, o4 + i);       // global_store_b128, TH=NT
  }

  // Scalar tail for n % 4 != 0 (no-op for the 4*4096*4096 reference shape).
  for (int j = numV4 * 4 + tid; j < n; j += stride) {
    out[j] = madq1(x[j], m, d, z);
  }
}

extern "C" void kernel_launch(void* const* d_in, const int* in_sizes, int n_in,
                              void* d_out, int out_size, void* d_ws, size_t ws_size,
                              hipStream_t stream) {
  const float* x = (const float*)d_in[0];
  const float* medians = (const float*)d_in[1];
  const float* deltas = (const float*)d_in[2];
  const float* zps = (const float*)d_in[3];
  float* out = (float*)d_out;

  const int n = in_sizes[0];
  const int numV4 = n / 4;

  int blocks = (numV4 + BLOCK * ITERS - 1) / (BLOCK * ITERS);
  if (blocks < 1) blocks = 1;

  madq_kernel<<<blocks, BLOCK, 0, stream>>>(x, medians, deltas, zps, out,
                                            numV4, n);
}